// LocalAttention_22840636080333
// MI455X (gfx1250) — compile-verified
//
#include <hip/hip_runtime.h>
#include <hip/hip_bf16.h>
#include <math.h>

// LocalAttention (Luong "general" score, local-p gaussian window), MI455X/gfx1250.
// N=64, L=1024, H=1024, f32. Bandwidth-bound: 256 MB enc stream dominates
// (~11 us floor at 23.3 TB/s); GEMMs use V_WMMA_F32_16X16X4_F32 with 4
// independent accumulator chains per wave.
//
//   qw  = output @ W_a                       (WMMA GEMM, 64x1024)
//   s   = qw . enc  per (n,l)                (1024 blocks, streams 256 MB)
//   a   = softmax(s); w = a * gauss(p_t)     (gauss==0.0f for |l-p_t|>=64)
//   ctx = sum_l w_l enc_l  (windowed)        (<=127 rows/batch, mostly L2-hot)
//   out = tanh([ctx, output] @ W_c^T)        (WMMA GEMM)

typedef __attribute__((ext_vector_type(2))) float v2f;
typedef __attribute__((ext_vector_type(8))) float v8f;

#define NB 64
#define LL 1024
#define HH 1024

// ---------------------------------------------------------------------------
// Kernel 1: qw[m,n] = sum_k output[m,k] * W_a[k,n]   (M=64, N=1024, K=1024)
// One wave per 16x64 strip: 4 N-tiles share one A-fragment -> 4 independent
// WMMA chains. A (16x4 f32): lanes 0-15 K={0,1}, lanes 16-31 K={2,3}, M=lane&15.
// B (4x16): VGPR c holds K=c+2*(lane>>4), N=lane&15. D: VGPR i -> M=i+8*(lane>>4).
// ---------------------------------------------------------------------------
__global__ __launch_bounds__(32) void qw_gemm(const float* __restrict__ A,
                                              const float* __restrict__ Wa,
                                              float* __restrict__ qw) {
  const int lane = threadIdx.x & 31;
  const int half = lane >> 4;
  const int row  = lane & 15;
  const int m0 = blockIdx.x * 16;   // batch tile  (grid.x = 4)
  const int n0 = blockIdx.y * 64;   // 4 N-tiles   (grid.y = 16)

  v8f acc[4] = {};
  const float* ap = A  + (size_t)(m0 + row) * HH + 2 * half;
  const float* bp = Wa + (size_t)(2 * half) * HH + n0 + row;
  for (int k = 0; k < HH; k += 4) {
    v2f a;
    a.x = ap[k + 0];
    a.y = ap[k + 1];
#pragma unroll
    for (int t = 0; t < 4; ++t) {
      v2f b;
      b.x = bp[(size_t)k * HH + t * 16];
      b.y = bp[(size_t)k * HH + HH + t * 16];
      acc[t] = __builtin_amdgcn_wmma_f32_16x16x4_f32(false, a, false, b,
                                                     (short)0, acc[t], false, false);
    }
  }
#pragma unroll
  for (int t = 0; t < 4; ++t)
#pragma unroll
    for (int i = 0; i < 8; ++i)
      qw[(size_t)(m0 + i + 8 * half) * HH + n0 + t * 16 + row] = acc[t][i];
}

// ---------------------------------------------------------------------------
// Kernel 2: scores[n,l] = <qw[n,:], enc[n,l,:]>.  Grid (64, 16), block 256
// (8 waves): 1024 blocks of memory-level parallelism for the 256 MB stream.
// qW row register-resident (32 regs/lane); wave shfl-reduction; one
// global_prefetch_b8 per row covers the next 4 KB row (128 B/lane stride).
// ---------------------------------------------------------------------------
__global__ __launch_bounds__(256) void scores_k(const float* __restrict__ enc,
                                                const float* __restrict__ qw,
                                                float* __restrict__ sc_g) {
  const int n    = blockIdx.x;
  const int wave = threadIdx.x >> 5;
  const int lane = threadIdx.x & 31;
  const float* encn = enc + (size_t)n * LL * HH;

  float q[32];
#pragma unroll
  for (int j = 0; j < 32; ++j) q[j] = qw[(size_t)n * HH + j * 32 + lane];

  const int l0 = blockIdx.y * 64 + wave * 8;
  for (int i = 0; i < 8; ++i) {
    const int l = l0 + i;
    const float* er = encn + (size_t)l * HH;
    if (i < 7) __builtin_prefetch(er + HH + lane * 32, 0, 0);  // next row
    float p = 0.f;
#pragma unroll
    for (int j = 0; j < 32; ++j) p = fmaf(q[j], er[j * 32 + lane], p);
#pragma unroll
    for (int m = 16; m >= 1; m >>= 1) p += __shfl_xor(p, m, 32);
    if (lane == 0) sc_g[(size_t)n * LL + l] = p;
  }
}

// ---------------------------------------------------------------------------
// Kernel 3: softmax + gaussian window + windowed context + concat.
// One block (1024 thr) per batch. Context only needs |l - p_t| <= 63
// (gauss underflows to exact f32 zero beyond that).
// ---------------------------------------------------------------------------
__global__ __launch_bounds__(1024) void softmax_ctx(const float* __restrict__ enc,
                                                    const float* __restrict__ outp,
                                                    const float* __restrict__ sc_g,
                                                    const int* __restrict__ src_len,
                                                    const int* __restrict__ tstep,
                                                    float* __restrict__ cat) {
  __shared__ float w[LL];
  __shared__ float red[32];
  __shared__ float s_max, s_sum;

  const int n    = blockIdx.x;
  const int tid  = threadIdx.x;
  const int wave = tid >> 5;
  const int lane = tid & 31;
  const float* encn = enc + (size_t)n * LL * HH;

  const float v = sc_g[(size_t)n * LL + tid];

  // workgroup max
  float mx = v;
#pragma unroll
  for (int m = 16; m >= 1; m >>= 1) mx = fmaxf(mx, __shfl_xor(mx, m, 32));
  if (lane == 0) red[wave] = mx;
  __syncthreads();
  if (tid < 32) {
    float t = red[tid];
#pragma unroll
    for (int m = 16; m >= 1; m >>= 1) t = fmaxf(t, __shfl_xor(t, m, 32));
    if (tid == 0) s_max = t;
  }
  __syncthreads();

  // workgroup sum of exp
  const float e = expf(v - s_max);
  float sm = e;
#pragma unroll
  for (int m = 16; m >= 1; m >>= 1) sm += __shfl_xor(sm, m, 32);
  if (lane == 0) red[wave] = sm;
  __syncthreads();
  if (tid < 32) {
    float t = red[tid];
#pragma unroll
    for (int m = 16; m >= 1; m >>= 1) t += __shfl_xor(t, m, 32);
    if (tid == 0) s_sum = t;
  }
  __syncthreads();

  int pt_i = src_len[n] - tstep[0];
  if (pt_i < -1) pt_i = -1;
  const float dlt = (float)tid - (float)pt_i;
  const float g   = expf(-(dlt * dlt) * (1.0f / 25.0f));  // 0.0f for |dlt|>=64
  w[tid] = (e / s_sum) * g;
  __syncthreads();

  int lo = pt_i - 63; if (lo < 0) lo = 0;
  int hi = pt_i + 63; if (hi > LL - 1) hi = LL - 1;
  float c = 0.f;
  for (int l = lo; l <= hi; ++l)
    c = fmaf(w[l], encn[(size_t)l * HH + tid], c);  // coalesced across block
  cat[(size_t)n * (2 * HH) + tid]      = c;
  cat[(size_t)n * (2 * HH) + HH + tid] = outp[(size_t)n * HH + tid];
}

// ---------------------------------------------------------------------------
// Kernel 4: out[b,h] = tanh( sum_d cat[b,d] * W_c[h,d] )  (M=h, N=b, K=2048).
// One wave per 16-row h-tile handles all 4 batch tiles: A=W_c fragment shared
// by 4 independent WMMA chains; both operands stream row-major.
// ---------------------------------------------------------------------------
__global__ __launch_bounds__(32) void out_gemm(const float* __restrict__ Wc,
                                               const float* __restrict__ cat,
                                               float* __restrict__ out) {
  const int lane = threadIdx.x & 31;
  const int half = lane >> 4;
  const int row  = lane & 15;
  const int m0 = blockIdx.x * 16;   // h tile (grid.x = 64)

  v8f acc[4] = {};
  const float* ap = Wc + (size_t)(m0 + row) * (2 * HH) + 2 * half;
  for (int k = 0; k < 2 * HH; k += 4) {
    v2f a;
    a.x = ap[k + 0];
    a.y = ap[k + 1];
#pragma unroll
    for (int t = 0; t < 4; ++t) {
      v2f b;
      const float* bp = cat + (size_t)(t * 16 + row) * (2 * HH) + k + 2 * half;
      b.x = bp[0];
      b.y = bp[1];
      acc[t] = __builtin_amdgcn_wmma_f32_16x16x4_f32(false, a, false, b,
                                                     (short)0, acc[t], false, false);
    }
  }
#pragma unroll
  for (int t = 0; t < 4; ++t)
#pragma unroll
    for (int i = 0; i < 8; ++i)
      out[(size_t)(t * 16 + row) * HH + m0 + i + 8 * half] = tanhf(acc[t][i]);
}

extern "C" void kernel_launch(void* const* d_in, const int* in_sizes, int n_in,
                              void* d_out, int out_size, void* d_ws, size_t ws_size,
                              hipStream_t stream) {
  const float* enc  = (const float*)d_in[0];  // (64,1024,1024)
  const float* outp = (const float*)d_in[1];  // (64,1,1024)
  const float* Wa   = (const float*)d_in[2];  // (1024,1024)
  const float* Wc   = (const float*)d_in[3];  // (1024,2048)
  const int*   srcl = (const int*)d_in[4];    // (64,)
  const int*   ts   = (const int*)d_in[5];    // scalar
  float* out = (float*)d_out;                 // (64,1,1024)

  float* qw   = (float*)d_ws;                 // 64*1024
  float* cat  = qw + NB * HH;                 // 64*2048
  float* sc_g = cat + NB * 2 * HH;            // 64*1024

  qw_gemm<<<dim3(NB / 16, HH / 64), 32, 0, stream>>>(outp, Wa, qw);
  scores_k<<<dim3(NB, LL / 64), 256, 0, stream>>>(enc, qw, sc_g);
  softmax_ctx<<<NB, 1024, 0, stream>>>(enc, outp, sc_g, srcl, ts, cat);
  out_gemm<<<HH / 16, 32, 0, stream>>>(Wc, cat, out);
}